// SIRD_72773925863765
// MI455X (gfx1250) — compile-verified
//
#include <hip/hip_runtime.h>
#include <hip/hip_bf16.h>

// SIRD RK4 batch integrator for MI455X (gfx1250).
// One trajectory per lane; 64 blocks x 32 threads = 64 wave32s (one per WGP).
// Reduced state: (S, I) recurrence; D accumulated from RK4 stage I-values.
// Output staged in padded LDS tiles, drained via CDNA5 async LDS->global DMA
// (ASYNCcnt) overlapped with compute of the next tile.

#define T_PTS 2048
#define TILE  32
#define ROWP  33   // LDS row pitch in float2 (pad to kill bank conflicts)

#if defined(__AMDGCN__) && defined(__gfx1250__) && \
    __has_builtin(__builtin_amdgcn_global_store_async_from_lds_b64)
#define USE_ASYNC 1
#else
#define USE_ASYNC 0
#endif

#if USE_ASYNC
typedef int v2i __attribute__((ext_vector_type(2)));
typedef __attribute__((address_space(1))) v2i as1_v2i;   // global <2 x i32>
typedef __attribute__((address_space(3))) v2i as3_v2i;   // LDS    <2 x i32>
#endif

__global__ __launch_bounds__(32, 1)
void sird_rk4_kernel(const float* __restrict__ alpha, float* __restrict__ out) {
  const int lane      = threadIdx.x;            // 0..31, trajectory within wave
  const int blockBase = blockIdx.x * 32;        // first trajectory of this wave
  const int traj      = blockBase + lane;

  const float beta  = alpha[3 * traj + 0];
  const float gamma = alpha[3 * traj + 1];
  const float mu    = alpha[3 * traj + 2];

  const float c   = beta * 1.0e-7f;             // beta / N
  const float d   = gamma + mu;
  const float h   = 0.125f;                     // dt = 1/SUBSTEPS
  const float h2  = 0.0625f;                    // dt/2
  const float h6  = h * (1.0f / 6.0f);          // dt/6
  const float dmu = h6 * mu;                    // D accumulation weight

  float S = 1.0e7f - 1.0f;
  float I = 1.0f;
  float D = 0.0f;

  // Double-buffered staging: [buf][trajectory_row * ROWP + time_in_tile]
  __shared__ float2 buf[2][32 * ROWP];          // ~16.9 KB of 320 KB LDS

  for (int tile = 0; tile < T_PTS / TILE; ++tile) {
    float2* b = &buf[tile & 1][0];

#if USE_ASYNC
    // Buffer (tile-2) reuse: with <=32 outstanding async stores, everything
    // older than the previous tile's 32 stores has retired (in-order).
    asm volatile("s_wait_asynccnt 32" ::: "memory");
#endif

#pragma unroll 1
    for (int tt = 0; tt < TILE; ++tt) {
      const int t = tile * TILE + tt;           // uniform across lanes
      if (t > 0) {
#pragma unroll
        for (int s = 0; s < 8; ++s) {
          // Stage 1 (at S, I)
          const float t1  = c * S * I;
          const float kI1 = __builtin_fmaf(-d, I, t1);
          const float S1  = __builtin_fmaf(-h2, t1, S);
          const float I1  = __builtin_fmaf(h2, kI1, I);
          // Stage 2 (at S1, I1)
          const float t2  = c * S1 * I1;
          const float kI2 = __builtin_fmaf(-d, I1, t2);
          const float S2  = __builtin_fmaf(-h2, t2, S);
          const float I2  = __builtin_fmaf(h2, kI2, I);
          // Stage 3 (at S2, I2)
          const float t3  = c * S2 * I2;
          const float kI3 = __builtin_fmaf(-d, I2, t3);
          const float S3  = __builtin_fmaf(-h, t3, S);
          const float I3  = __builtin_fmaf(h, kI3, I);
          // Stage 4 (at S3, I3)
          const float t4  = c * S3 * I3;
          const float kI4 = __builtin_fmaf(-d, I3, t4);
          // Combine
          const float sumT = __builtin_fmaf(2.0f, t2 + t3, t1 + t4);
          const float sumI = __builtin_fmaf(2.0f, kI2 + kI3, kI1 + kI4);
          const float sumD = __builtin_fmaf(2.0f, I1 + I2, I + I3);
          S = __builtin_fmaf(-h6, sumT, S);
          I = __builtin_fmaf(h6, sumI, I);
          D = __builtin_fmaf(dmu, sumD, D);
        }
      }
      // Lane (= trajectory) writes its (I, D) for this time step.
      // Offset lane*ROWP*8 + tt*8: padded pitch -> all 64 banks distinct.
      b[lane * ROWP + tt] = make_float2(I, D);
    }

#if USE_ASYNC
    // Publish LDS writes to the async DMA engine, then fire 32 row-stores:
    // row r = trajectory blockBase+r, lane l covers time tile*32+l (8 B/lane,
    // 256 B contiguous per instruction -> coalesced), overlapped with the
    // next tile's compute via ASYNCcnt.
    asm volatile("s_wait_dscnt 0" ::: "memory");
#pragma unroll 1
    for (int r = 0; r < 32; ++r) {
      float* g = out + ((size_t)(blockBase + r) * T_PTS + tile * TILE + lane) * 2;
      v2i* gv = reinterpret_cast<v2i*>(g);
      v2i* lv = reinterpret_cast<v2i*>(&b[r * ROWP + lane]);
      __builtin_amdgcn_global_store_async_from_lds_b64(
          (as1_v2i*)gv, (as3_v2i*)lv, /*ioffset=*/0, /*cpol=*/0);
    }
#else
    __syncthreads();
#pragma unroll 1
    for (int r = 0; r < 32; ++r) {
      const float2 v = b[r * ROWP + lane];
      reinterpret_cast<float2*>(out)[(size_t)(blockBase + r) * T_PTS +
                                     tile * TILE + lane] = v;
    }
    __syncthreads();
#endif
  }

#if USE_ASYNC
  asm volatile("s_wait_asynccnt 0" ::: "memory");
#endif
}

extern "C" void kernel_launch(void* const* d_in, const int* in_sizes, int n_in,
                              void* d_out, int out_size, void* d_ws, size_t ws_size,
                              hipStream_t stream) {
  const float* alpha = (const float*)d_in[0];   // [2048, 3] f32
  float* out = (float*)d_out;                   // [2048, 2048, 2] f32
  (void)in_sizes; (void)n_in; (void)out_size; (void)d_ws; (void)ws_size;
  sird_rk4_kernel<<<dim3(T_PTS / 32), dim3(32), 0, stream>>>(alpha, out);
}